// GlobalNet_58042188038823
// MI455X (gfx1250) — compile-verified
//
#include <hip/hip_runtime.h>
#include <hip/hip_bf16.h>
#include <math.h>

// ---------------------------------------------------------------------------
// GlobalNet mesh-conv network for MI455X (gfx1250), wave32 + WMMA f32 16x16x4.
// Activations kept in [b][e][c] layout so neighbor gathers are b128-contiguous.
// ---------------------------------------------------------------------------

typedef __attribute__((ext_vector_type(2))) float v2f;
typedef __attribute__((ext_vector_type(8))) float v8f;

static constexpr int BB = 8;
static constexpr int EE = 65536;
static constexpr int FC_N = 2500;
static constexpr int OUTM = 642 * 3;   // 1926
static constexpr float EPS = 1e-5f;

// ---------------------------------------------------------------------------
// prep: zero stat accumulators + pooled accumulator, build zero-padded weights
// ---------------------------------------------------------------------------
__global__ void prep_kernel(const float* __restrict__ w0, const float* __restrict__ w1,
                            const float* __restrict__ w2, const float* __restrict__ w3,
                            float* __restrict__ wp0, float* __restrict__ wp1,
                            float* __restrict__ wp2, float* __restrict__ wp3,
                            float* __restrict__ zero_region, int zero_count) {
  int t = blockIdx.x * blockDim.x + threadIdx.x;
  int n = gridDim.x * blockDim.x;
  for (int i = t; i < zero_count; i += n) zero_region[i] = 0.f;
  // layer0: CO=16, K=25 -> KP=28 (zero pad)
  for (int i = t; i < 16 * 28; i += n) {
    int o = i / 28, k = i % 28;
    wp0[i] = (k < 25) ? w0[o * 25 + k] : 0.f;
  }
  for (int i = t; i < 32 * 80; i += n)  wp1[i] = w1[i];
  for (int i = t; i < 64 * 160; i += n) wp2[i] = w2[i];
  for (int i = t; i < 64 * 320; i += n) wp3[i] = w3[i];
}

// ---------------------------------------------------------------------------
// transpose x (B,5,E) -> xt (B,E,5)
// ---------------------------------------------------------------------------
__global__ __launch_bounds__(256) void transpose_kernel(const float* __restrict__ x,
                                                        float* __restrict__ xt) {
  size_t t = (size_t)blockIdx.x * 256 + threadIdx.x;  // over B*E
  if (t >= (size_t)BB * EE) return;
  size_t b = t / EE, e = t % EE;
#pragma unroll
  for (int c = 0; c < 5; ++c)
    xt[t * 5 + c] = x[(b * 5 + c) * EE + e];
}

// ---------------------------------------------------------------------------
// mesh conv layer: in [b][e][CI] (pre-BN), out [b][e][CO] (pre-BN),
// accumulates per-channel sum / sum^2 for the next BN.
// One block = 32 e-columns of one batch row; 8 waves of 32 (wave32).
// ---------------------------------------------------------------------------
template <int CI, int CO, int KP>
__global__ __launch_bounds__(256) void conv_kernel(
    const float* __restrict__ xin, const int* __restrict__ gemm,
    const float* __restrict__ wpad, const float* __restrict__ bias,
    const float* __restrict__ scale, const float* __restrict__ shift,
    float* __restrict__ hout, float* __restrict__ gsum, float* __restrict__ gsq,
    int applyBn) {
  constexpr int NE = 32;                 // e-tile width
  constexpr int K = CI * 5;              // true reduction length
  constexpr int SLOT = NE + 1;           // LDS pad to dodge bank conflicts
  constexpr int SSTR = CO + 4;           // staging row stride (16B aligned)
  constexpr int STG = NE * SSTR;         // output staging size
  constexpr int FSZ = (KP * NE > STG) ? KP * NE : STG;
  constexpr int MT = CO / 16;
  constexpr int NT = NE / 16;
  constexpr int TILES = MT * NT;         // <= 8 always

  __shared__ float s_nb[5 * CI * SLOT];  // [slot][c][e]
  __shared__ __align__(16) float s_f[FSZ];  // feature tile [KP][NE], reused as staging [NE][SSTR]
  __shared__ int   s_idx[NE * 4];
  __shared__ float s_scale[CI], s_shift[CI], s_bias[CO], s_sum[CO], s_sq[CO];

  const int tid = threadIdx.x;
  const int ntiles = EE / NE;
  const int b = blockIdx.x / ntiles;
  const int e0 = (blockIdx.x % ntiles) * NE;

  if (tid < NE * 4)
    s_idx[tid] = gemm[((size_t)b * EE + e0 + (tid >> 2)) * 4 + (tid & 3)];
  if (tid < CI && applyBn) { s_scale[tid] = scale[tid]; s_shift[tid] = shift[tid]; }
  if (tid < CO) { s_bias[tid] = bias[tid]; s_sum[tid] = 0.f; s_sq[tid] = 0.f; }
  __syncthreads();

  // ---- Phase A: gather center + 4 neighbors, apply previous BN+ReLU -------
  const float* xb = xin + (size_t)b * EE * CI;
  if constexpr (CI % 4 == 0) {
    constexpr int C4 = CI / 4;           // b128 gathers
    for (int it = tid; it < 5 * NE * C4; it += 256) {
      int s = it / (NE * C4);
      int r = it % (NE * C4);
      int e = r / C4;
      int c4 = r % C4;
      int idx = (s == 0) ? (e0 + e) : s_idx[e * 4 + (s - 1)];
      float4 v = *(const float4*)(xb + (size_t)idx * CI + c4 * 4);
      if (applyBn) {
        int cb = c4 * 4;
        v.x = fmaxf(v.x * s_scale[cb + 0] + s_shift[cb + 0], 0.f);
        v.y = fmaxf(v.y * s_scale[cb + 1] + s_shift[cb + 1], 0.f);
        v.z = fmaxf(v.z * s_scale[cb + 2] + s_shift[cb + 2], 0.f);
        v.w = fmaxf(v.w * s_scale[cb + 3] + s_shift[cb + 3], 0.f);
      }
      int a = (s * CI + c4 * 4) * SLOT + e;
      s_nb[a] = v.x;
      s_nb[a + SLOT] = v.y;
      s_nb[a + 2 * SLOT] = v.z;
      s_nb[a + 3 * SLOT] = v.w;
    }
  } else {
    for (int it = tid; it < 5 * NE * CI; it += 256) {
      int s = it / (NE * CI);
      int r = it % (NE * CI);
      int e = r / CI;
      int c = r % CI;
      int idx = (s == 0) ? (e0 + e) : s_idx[e * 4 + (s - 1)];
      float v = xb[(size_t)idx * CI + c];
      if (applyBn) v = fmaxf(v * s_scale[c] + s_shift[c], 0.f);
      s_nb[(s * CI + c) * SLOT + e] = v;
    }
  }
  __syncthreads();

  // ---- Phase B: build feature tile f[K][NE] -------------------------------
  for (int it = tid; it < CI * NE; it += 256) {
    int c = it / NE, e = it % NE;
    float ctr = s_nb[(0 * CI + c) * SLOT + e];
    float n1  = s_nb[(1 * CI + c) * SLOT + e];
    float n2  = s_nb[(2 * CI + c) * SLOT + e];
    float n3  = s_nb[(3 * CI + c) * SLOT + e];
    float n4  = s_nb[(4 * CI + c) * SLOT + e];
    int kb = c * 5;
    s_f[(kb + 0) * NE + e] = ctr;
    s_f[(kb + 1) * NE + e] = n1 + n3;
    s_f[(kb + 2) * NE + e] = n2 + n4;
    s_f[(kb + 3) * NE + e] = fabsf(n1 - n3);
    s_f[(kb + 4) * NE + e] = fabsf(n2 - n4);
  }
  if (KP > K) {  // zero-pad rows K..KP-1 (layer 0 only)
    for (int it = tid; it < (KP - K) * NE; it += 256) s_f[K * NE + it] = 0.f;
  }
  __syncthreads();

  // ---- WMMA: D(16x16) = Wpad(16xKP) * F(KPx16) ----------------------------
  const int wave = tid >> 5;
  const int lane = tid & 31;
  const int half = lane >> 4;
  const int lid  = lane & 15;
  int mt = 0, nt = 0;
  v8f acc = {0.f, 0.f, 0.f, 0.f, 0.f, 0.f, 0.f, 0.f};
  if (wave < TILES) {
    mt = wave / NT;
    nt = wave % NT;
    const int m = mt * 16 + lid;
    const int n = nt * 16 + lid;
    const float* wrow = wpad + (size_t)m * KP;  // 8B-aligned (KP even)
#pragma unroll 4
    for (int k = 0; k < KP; k += 4) {
      // A frag (16x4 f32): VGPR0 = K+0 | K+2, VGPR1 = K+1 | K+3 per lane-half
      v2f a = *(const v2f*)(wrow + k + 2 * half);
      // B frag (4x16 f32): rows striped across lanes, same K split
      v2f bf;
      bf.x = s_f[(k + 2 * half + 0) * NE + n];
      bf.y = s_f[(k + 2 * half + 1) * NE + n];
      acc = __builtin_amdgcn_wmma_f32_16x16x4_f32(false, a, false, bf,
                                                  (short)0, acc, false, false);
    }
  }
  __syncthreads();  // all f-tile reads done; reuse s_f as staging [NE][SSTR]

  if (wave < TILES) {
#pragma unroll
    for (int r = 0; r < 8; ++r) {
      int mm = mt * 16 + r + 8 * half;   // C/D layout: VGPR r holds M=r (lanes 0-15), M=r+8
      int nn = nt * 16 + lid;
      s_f[nn * SSTR + mm] = acc[r] + s_bias[mm];
    }
  }
  __syncthreads();

  // ---- coalesced b128 store + per-channel stats ---------------------------
  constexpr int C4O = CO / 4;            // divides 256 -> channel group fixed
  const int c4o = tid % C4O;
  float ps[4] = {0.f, 0.f, 0.f, 0.f};
  float pq[4] = {0.f, 0.f, 0.f, 0.f};
  float* ob = hout + ((size_t)b * EE + e0) * CO;
  for (int it = tid; it < NE * C4O; it += 256) {
    int e = it / C4O;
    float4 v = *(const float4*)&s_f[e * SSTR + c4o * 4];
    *(float4*)(ob + (size_t)e * CO + c4o * 4) = v;
    ps[0] += v.x; pq[0] += v.x * v.x;
    ps[1] += v.y; pq[1] += v.y * v.y;
    ps[2] += v.z; pq[2] += v.z * v.z;
    ps[3] += v.w; pq[3] += v.w * v.w;
  }
#pragma unroll
  for (int j = 0; j < 4; ++j) {
    atomicAdd(&s_sum[c4o * 4 + j], ps[j]);
    atomicAdd(&s_sq[c4o * 4 + j], pq[j]);
  }
  __syncthreads();
  if (tid < CO) {
    atomicAdd(&gsum[tid], s_sum[tid]);
    atomicAdd(&gsq[tid], s_sq[tid]);
  }
}

// ---------------------------------------------------------------------------
// BN stats -> scale/shift
// ---------------------------------------------------------------------------
__global__ void stats_kernel(const float* __restrict__ gsum, const float* __restrict__ gsq,
                             const float* __restrict__ g, const float* __restrict__ beta,
                             float* __restrict__ scale, float* __restrict__ shift, int co) {
  int c = threadIdx.x;
  if (c >= co) return;
  const float inv = 1.f / (float)((size_t)BB * EE);
  float mean = gsum[c] * inv;
  float var = gsq[c] * inv - mean * mean;
  float sc = g[c] * rsqrtf(var + EPS);
  scale[c] = sc;
  shift[c] = beta[c] - mean * sc;
}

// ---------------------------------------------------------------------------
// pool: relu(bn(h3)) mean over E -> pooled_sum[b][64] (divide later)
// ---------------------------------------------------------------------------
__global__ __launch_bounds__(256) void pool_kernel(const float* __restrict__ h,
                                                   const float* __restrict__ scale,
                                                   const float* __restrict__ shift,
                                                   float* __restrict__ pooled) {
  __shared__ float s_acc[64];
  const int b = blockIdx.y;
  const int e0 = blockIdx.x * 256;
  const int tid = threadIdx.x;
  if (tid < 64) s_acc[tid] = 0.f;
  __syncthreads();
  const int c4 = tid & 15;               // 16 channel-quads; fixed per thread
  const float4 sc = *(const float4*)(scale + c4 * 4);
  const float4 sh = *(const float4*)(shift + c4 * 4);
  float ps[4] = {0.f, 0.f, 0.f, 0.f};
  const float* hb = h + (((size_t)b * EE) + e0) * 64;
  for (int it = tid; it < 256 * 16; it += 256) {
    int e = it >> 4;                     // c4 stays == tid&15
    float4 v = *(const float4*)(hb + (size_t)e * 64 + c4 * 4);
    ps[0] += fmaxf(v.x * sc.x + sh.x, 0.f);
    ps[1] += fmaxf(v.y * sc.y + sh.y, 0.f);
    ps[2] += fmaxf(v.z * sc.z + sh.z, 0.f);
    ps[3] += fmaxf(v.w * sc.w + sh.w, 0.f);
  }
#pragma unroll
  for (int j = 0; j < 4; ++j) atomicAdd(&s_acc[c4 * 4 + j], ps[j]);
  __syncthreads();
  if (tid < 64) atomicAdd(&pooled[b * 64 + tid], s_acc[tid]);
}

// ---------------------------------------------------------------------------
// FC head
// ---------------------------------------------------------------------------
__global__ void fc1_kernel(const float* __restrict__ pooled, const float* __restrict__ dw1,
                           const float* __restrict__ db1, float* __restrict__ h1) {
  int n = blockIdx.x * blockDim.x + threadIdx.x;
  if (n >= FC_N) return;
  const float invE = 1.f / (float)EE;
  float acc[BB];
#pragma unroll
  for (int b = 0; b < BB; ++b) acc[b] = db1[n];
  for (int c = 0; c < 64; ++c) {
    float w = dw1[n * 64 + c];
#pragma unroll
    for (int b = 0; b < BB; ++b) acc[b] += (pooled[b * 64 + c] * invE) * w;
  }
#pragma unroll
  for (int b = 0; b < BB; ++b) h1[b * FC_N + n] = acc[b] > 0.f ? acc[b] : 0.f;
}

__global__ void fc2_kernel(const float* __restrict__ h1, const float* __restrict__ dw2,
                           const float* __restrict__ db2, const float* __restrict__ ov,
                           float* __restrict__ out) {
  int m = blockIdx.x * blockDim.x + threadIdx.x;
  if (m >= OUTM) return;
  float acc[BB];
#pragma unroll
  for (int b = 0; b < BB; ++b) acc[b] = db2[m];
  const float4* wrow4 = (const float4*)(dw2 + (size_t)m * FC_N);  // 10000B rows: 16B aligned
  for (int n4 = 0; n4 < FC_N / 4; ++n4) {
    float4 w4 = wrow4[n4];
#pragma unroll
    for (int b = 0; b < BB; ++b) {
      float4 h4 = *(const float4*)(h1 + (size_t)b * FC_N + n4 * 4);
      acc[b] += h4.x * w4.x + h4.y * w4.y + h4.z * w4.z + h4.w * w4.w;
    }
  }
#pragma unroll
  for (int b = 0; b < BB; ++b) out[b * OUTM + m] = acc[b] + ov[b * OUTM + m];
}

// ---------------------------------------------------------------------------
// launch
// ---------------------------------------------------------------------------
extern "C" void kernel_launch(void* const* d_in, const int* in_sizes, int n_in,
                              void* d_out, int out_size, void* d_ws, size_t ws_size,
                              hipStream_t stream) {
  const float* x    = (const float*)d_in[0];
  const int*   gemm = (const int*)d_in[1];
  const float* ov   = (const float*)d_in[2];
  const float* w[4], *bi[4], *g[4], *be[4];
  for (int i = 0; i < 4; ++i) {
    w[i]  = (const float*)d_in[3 + 4 * i + 0];
    bi[i] = (const float*)d_in[3 + 4 * i + 1];
    g[i]  = (const float*)d_in[3 + 4 * i + 2];
    be[i] = (const float*)d_in[3 + 4 * i + 3];
  }
  const float* dw1 = (const float*)d_in[19];
  const float* db1 = (const float*)d_in[20];
  const float* dw2 = (const float*)d_in[21];
  const float* db2 = (const float*)d_in[22];
  float* out = (float*)d_out;
  float* ws = (float*)d_ws;

  // workspace layout (all float offsets are multiples of 4 -> 16B alignment)
  size_t off = 0;
  float* xt   = ws + off; off += (size_t)BB * EE * 5;
  float* ping = ws + off; off += (size_t)BB * EE * 64;
  float* pong = ws + off; off += (size_t)BB * EE * 64;
  float* wp0  = ws + off; off += 16 * 28;
  float* wp1  = ws + off; off += 32 * 80;
  float* wp2  = ws + off; off += 64 * 160;
  float* wp3  = ws + off; off += 64 * 320;
  float* zr   = ws + off;                 // zero region start
  float* gsum = ws + off; off += 4 * 64;  // per-layer sums
  float* gsq  = ws + off; off += 4 * 64;
  float* pooled = ws + off; off += BB * 64;
  int zero_count = (int)(4 * 64 + 4 * 64 + BB * 64);
  float* scl  = ws + off; off += 4 * 64;
  float* shf  = ws + off; off += 4 * 64;
  float* h1   = ws + off; off += (size_t)BB * FC_N;

  prep_kernel<<<64, 256, 0, stream>>>(w[0], w[1], w[2], w[3], wp0, wp1, wp2, wp3,
                                      zr, zero_count);
  transpose_kernel<<<(BB * EE + 255) / 256, 256, 0, stream>>>(x, xt);

  const int grid = BB * EE / 32;

  conv_kernel<5, 16, 28><<<grid, 256, 0, stream>>>(
      xt, gemm, wp0, bi[0], scl, shf, ping, gsum + 0, gsq + 0, 0);
  stats_kernel<<<1, 64, 0, stream>>>(gsum + 0, gsq + 0, g[0], be[0], scl + 0, shf + 0, 16);

  conv_kernel<16, 32, 80><<<grid, 256, 0, stream>>>(
      ping, gemm, wp1, bi[1], scl + 0, shf + 0, pong, gsum + 64, gsq + 64, 1);
  stats_kernel<<<1, 64, 0, stream>>>(gsum + 64, gsq + 64, g[1], be[1], scl + 64, shf + 64, 32);

  conv_kernel<32, 64, 160><<<grid, 256, 0, stream>>>(
      pong, gemm, wp2, bi[2], scl + 64, shf + 64, ping, gsum + 128, gsq + 128, 1);
  stats_kernel<<<1, 64, 0, stream>>>(gsum + 128, gsq + 128, g[2], be[2], scl + 128, shf + 128, 64);

  conv_kernel<64, 64, 320><<<grid, 256, 0, stream>>>(
      ping, gemm, wp3, bi[3], scl + 128, shf + 128, pong, gsum + 192, gsq + 192, 1);
  stats_kernel<<<1, 64, 0, stream>>>(gsum + 192, gsq + 192, g[3], be[3], scl + 192, shf + 192, 64);

  dim3 pg(EE / 256, BB);
  pool_kernel<<<pg, 256, 0, stream>>>(pong, scl + 192, shf + 192, pooled);
  fc1_kernel<<<(FC_N + 255) / 256, 256, 0, stream>>>(pooled, dw1, db1, h1);
  fc2_kernel<<<(OUTM + 255) / 256, 256, 0, stream>>>(h1, dw2, db2, ov, out);
}